// RankingLoss_27152783245561
// MI455X (gfx1250) — compile-verified
//
#include <hip/hip_runtime.h>
#include <cstdint>
#include <cfloat>
#include <cmath>

// Problem geometry (fixed by the reference): B=16, C=8, H=W=512.
#define PLANES       128
#define PLANE_ELEMS  (512 * 512)                 // 262144
#define CHUNKS       16
#define TPB          256
#define VEC          4
#define CHUNK_ELEMS  (PLANE_ELEMS / CHUNKS)      // 16384
#define ITERS        (CHUNK_ELEMS / (TPB * VEC)) // 16
#define NBUF         4                           // LDS ring depth (WAR safety distance)

typedef __attribute__((ext_vector_type(4))) float v4f;
// Must match the builtin's parameter pointee exactly: GCC-style int 4-vector.
typedef int v4i __attribute__((vector_size(4 * sizeof(int))));

// ---- CDNA5 async global->LDS copy (per-lane 16B), builtin if present else ISA asm ----
__device__ __forceinline__ void async_load_b128(const float* g, v4f* lds) {
#if __has_builtin(__builtin_amdgcn_global_load_async_to_lds_b128)
  __builtin_amdgcn_global_load_async_to_lds_b128(
      (v4i*)g, (v4i*)lds, /*imm_off=*/0, /*cpol=*/0);
#else
  const uint32_t lds_byte_off = (uint32_t)(uintptr_t)lds;  // low 32b of flat == LDS offset
  asm volatile("global_load_async_to_lds_b128 %0, %1, off"
               :: "v"(lds_byte_off), "v"(g)
               : "memory");
#endif
}

__device__ __forceinline__ void wait_asynccnt_le2() {
#if __has_builtin(__builtin_amdgcn_s_wait_asynccnt)
  __builtin_amdgcn_s_wait_asynccnt(2);
#else
  asm volatile("s_wait_asynccnt 2" ::: "memory");
#endif
}

__device__ __forceinline__ void wait_asynccnt_le0() {
#if __has_builtin(__builtin_amdgcn_s_wait_asynccnt)
  __builtin_amdgcn_s_wait_asynccnt(0);
#else
  asm volatile("s_wait_asynccnt 0" ::: "memory");
#endif
}

// Kernel 1: per-(plane,chunk) masked min/max over raw pred (sigmoid is monotone).
__global__ __launch_bounds__(TPB) void rank_partial_kernel(
    const float* __restrict__ pred, const float* __restrict__ tru,
    float* __restrict__ ws_min, float* __restrict__ ws_max) {
  __shared__ v4f sp[NBUF][TPB];        // staged pred tiles
  __shared__ v4f st[NBUF][TPB];        // staged true tiles
  __shared__ float red_min[TPB / 32];
  __shared__ float red_max[TPB / 32];

  const int t     = threadIdx.x;
  const int plane = blockIdx.x;        // 0..127
  const int chunk = blockIdx.y;        // 0..CHUNKS-1
  const long base = (long)plane * PLANE_ELEMS + (long)chunk * CHUNK_ELEMS;
  const float* gp = pred + base + t * VEC;
  const float* gt = tru  + base + t * VEC;

  float posMin =  INFINITY;   // min pred where true > 0.5  (+inf == "no positives")
  float negMax = -INFINITY;   // max pred where true < 0.3  (-inf == "no negatives")

  // Prime the pipeline: stage iteration 0 into ring slot 0.
  async_load_b128(gp, &sp[0][t]);
  async_load_b128(gt, &st[0][t]);

  for (int i = 0; i < ITERS; ++i) {
    if (i + 1 < ITERS) {
      const int adv = (i + 1) * TPB * VEC;
      async_load_b128(gp + adv, &sp[(i + 1) & (NBUF - 1)][t]);
      async_load_b128(gt + adv, &st[(i + 1) & (NBUF - 1)][t]);
      wait_asynccnt_le2();     // in-order completion: current tile's 2 copies are done
    } else {
      wait_asynccnt_le0();
    }
    const v4f p  = sp[i & (NBUF - 1)][t];   // ds_load_b128 (own slot; no barrier needed)
    const v4f tv = st[i & (NBUF - 1)][t];
#pragma unroll
    for (int k = 0; k < VEC; ++k) {
      posMin = (tv[k] > 0.5f) ? fminf(posMin, p[k]) : posMin;
      negMax = (tv[k] < 0.3f) ? fmaxf(negMax, p[k]) : negMax;
    }
  }

  // wave32 butterfly reduction
#pragma unroll
  for (int off = 16; off > 0; off >>= 1) {
    posMin = fminf(posMin, __shfl_xor(posMin, off, 32));
    negMax = fmaxf(negMax, __shfl_xor(negMax, off, 32));
  }
  const int wave = t >> 5;
  const int lane = t & 31;
  if (lane == 0) { red_min[wave] = posMin; red_max[wave] = negMax; }
  __syncthreads();
  if (t == 0) {
    float m = red_min[0], M = red_max[0];
#pragma unroll
    for (int w = 1; w < TPB / 32; ++w) {
      m = fminf(m, red_min[w]);
      M = fmaxf(M, red_max[w]);
    }
    ws_min[plane * CHUNKS + chunk] = m;
    ws_max[plane * CHUNKS + chunk] = M;
  }
}

// Kernel 2: fold chunk partials per plane, apply sigmoid/exp semantics, sum, scale.
__global__ __launch_bounds__(PLANES) void rank_final_kernel(
    const float* __restrict__ ws_min, const float* __restrict__ ws_max,
    float* __restrict__ out) {
  __shared__ float acc[PLANES];
  const int t = threadIdx.x;  // one thread per plane

  float m =  INFINITY;
  float M = -INFINITY;
#pragma unroll
  for (int c = 0; c < CHUNKS; ++c) {
    m = fminf(m, ws_min[t * CHUNKS + c]);
    M = fmaxf(M, ws_max[t * CHUNKS + c]);
  }
  const bool hasPos = (m <  INFINITY);
  const bool hasNeg = (M > -INFINITY);
  // Reference semantics: where() applied AFTER sigmoid with +/-FLT_MAX fill.
  const float xpos = 1.0f / (1.0f + expf(-m));            // valid only if hasPos
  const float xneg = hasNeg ? 1.0f / (1.0f + expf(-M)) : -FLT_MAX;
  const float val  = hasPos ? expf(xneg - xpos) : 0.0f;   // exp(-FLT_MAX - x) -> 0

  acc[t] = val;
  __syncthreads();
#pragma unroll
  for (int off = PLANES / 2; off > 0; off >>= 1) {
    if (t < off) acc[t] += acc[t + off];
    __syncthreads();
  }
  if (t == 0) out[0] = acc[0] * (1.0f / (float)PLANES);
}

extern "C" void kernel_launch(void* const* d_in, const int* in_sizes, int n_in,
                              void* d_out, int out_size, void* d_ws, size_t ws_size,
                              hipStream_t stream) {
  (void)in_sizes; (void)n_in; (void)out_size; (void)ws_size;
  const float* pred = (const float*)d_in[0];
  const float* tru  = (const float*)d_in[1];
  float* ws_min = (float*)d_ws;                    // PLANES*CHUNKS floats
  float* ws_max = ws_min + PLANES * CHUNKS;        // PLANES*CHUNKS floats (16 KB total)

  dim3 grid(PLANES, CHUNKS);
  rank_partial_kernel<<<grid, TPB, 0, stream>>>(pred, tru, ws_min, ws_max);
  rank_final_kernel<<<1, PLANES, 0, stream>>>(ws_min, ws_max, (float*)d_out);
}